// LogEig_9423158247636
// MI455X (gfx1250) — compile-verified
//
#include <hip/hip_runtime.h>
#include <math.h>

// ---------------------------------------------------------------------------
// Batched SPD matrix logarithm, MI455X (gfx1250).
//   logm(A) = -2^k * sum_{j=1..8} E^j / j + log(t) I,
//   E = I - (A/t)^(1/2^k), k = 6, t = ||A||_1 >= lambda_max.
// Square roots via inverse-free coupled Newton-Schulz (GEMM-only).
// One workgroup (8 wave32s) per matrix; all 4 working 128x128 f32 buffers
// live in LDS (~264 KB of the 320 KB WGP pool). GEMMs use
// V_WMMA_F32_16X16X4_F32.
// ---------------------------------------------------------------------------

typedef float v2f __attribute__((ext_vector_type(2)));
typedef float v8f __attribute__((ext_vector_type(8)));

#define MATN   128
#define LDSS   132          // padded LDS row stride (floats): banks 4r -> no conflicts
#define NLEV   6            // number of repeated square roots (2^6 = 64)

// C = alpha * A*B + diag * I    (all pointers into LDS, 128x128, stride LDSS)
// 8 waves: wave (r,c) with r=wave>>1, c=wave&1 owns M-tiles {2r,2r+1} x N-tiles {4c..4c+3}.
__device__ __forceinline__ void gemm128(float* __restrict__ C,
                                        const float* __restrict__ A,
                                        const float* __restrict__ B,
                                        float alpha, float diag, int tid)
{
    const int lane = tid & 31;
    const int wave = tid >> 5;
    const int l15  = lane & 15;
    const int hi   = lane >> 4;          // 0: K{0,1} / rows M, 1: K{2,3} / rows M+8
    const int mt0  = (wave >> 1) * 2;    // first of 2 M-tiles (tiles of 16 rows)
    const int nt0  = (wave & 1) * 4;     // first of 4 N-tiles

    v8f acc[2][4];
#pragma unroll
    for (int i = 0; i < 2; ++i)
#pragma unroll
        for (int j = 0; j < 4; ++j)
            acc[i][j] = (v8f){0.f, 0.f, 0.f, 0.f, 0.f, 0.f, 0.f, 0.f};

    for (int k0 = 0; k0 < MATN; k0 += 4) {
        // A fragments: lane -> row (mt*16 + l15), columns k0+2*hi, k0+2*hi+1
        v2f afrag[2];
#pragma unroll
        for (int i = 0; i < 2; ++i) {
            const float* ap = A + ((mt0 + i) * 16 + l15) * LDSS + k0 + 2 * hi;
            afrag[i].x = ap[0];
            afrag[i].y = ap[1];
        }
        // B fragments: lane -> col (nt*16 + l15), rows k0+2*hi, k0+2*hi+1
        v2f bfrag[4];
#pragma unroll
        for (int j = 0; j < 4; ++j) {
            const float* bp = B + (k0 + 2 * hi) * LDSS + (nt0 + j) * 16 + l15;
            bfrag[j].x = bp[0];
            bfrag[j].y = bp[LDSS];
        }
#pragma unroll
        for (int i = 0; i < 2; ++i)
#pragma unroll
            for (int j = 0; j < 4; ++j)
                acc[i][j] = __builtin_amdgcn_wmma_f32_16x16x4_f32(
                    /*neg_a=*/false, afrag[i],
                    /*neg_b=*/false, bfrag[j],
                    /*c_mod=*/(short)0, acc[i][j],
                    /*reuse_a=*/false, /*reuse_b=*/false);
    }

    // C/D layout: VGPR r -> row r (lanes 0-15) or r+8 (lanes 16-31), col = l15
#pragma unroll
    for (int i = 0; i < 2; ++i)
#pragma unroll
        for (int j = 0; j < 4; ++j) {
            const int rowbase = (mt0 + i) * 16 + hi * 8;
            const int col     = (nt0 + j) * 16 + l15;
#pragma unroll
            for (int r = 0; r < 8; ++r) {
                const int row = rowbase + r;
                float v = alpha * acc[i][j][r];
                if (row == col) v += diag;
                C[row * LDSS + col] = v;
            }
        }
}

__global__ __launch_bounds__(256) void logm_spd_kernel(const float* __restrict__ in,
                                                       float* __restrict__ out)
{
    __shared__ float bufY[MATN * LDSS];
    __shared__ float bufZ[MATN * LDSS];
    __shared__ float bufT[MATN * LDSS];
    __shared__ float bufW[MATN * LDSS];
    __shared__ float red[MATN];
    __shared__ float snorm;

    const int    tid  = threadIdx.x;
    const size_t base = (size_t)blockIdx.x * MATN * MATN;

    // ---- load A into LDS (float4, coalesced) ----
    for (int idx = tid * 4; idx < MATN * MATN; idx += 256 * 4) {
        const float4 v = *(const float4*)(in + base + idx);
        const int row = idx >> 7, col = idx & 127;
        *(float4*)(bufY + row * LDSS + col) = v;
    }
    __syncthreads();

    // ---- t = ||A||_1 (max column abs-sum) >= lambda_max ----
    if (tid < MATN) {
        float s = 0.f;
        for (int i = 0; i < MATN; ++i) s += fabsf(bufY[i * LDSS + tid]);
        red[tid] = s;
    }
    __syncthreads();
    if (tid == 0) {
        float m = red[0];
        for (int i = 1; i < MATN; ++i) m = fmaxf(m, red[i]);
        snorm = m;
    }
    __syncthreads();
    const float t     = snorm;
    const float inv_t = 1.0f / t;

    // ---- Y = A/t, Z = I ----
    for (int idx = tid; idx < MATN * MATN; idx += 256) {
        const int row = idx >> 7, col = idx & 127;
        bufY[row * LDSS + col] *= inv_t;
        bufZ[row * LDSS + col] = (row == col) ? 1.0f : 0.0f;
    }
    __syncthreads();

    float* Y = bufY;
    float* Z = bufZ;
    float* T = bufT;
    float* W = bufW;

    // ---- k repeated square roots: coupled Newton-Schulz ----
    const int iters[NLEV] = {14, 10, 8, 6, 5, 4};
    for (int lvl = 0; lvl < NLEV; ++lvl) {
        for (int it = 0; it < iters[lvl]; ++it) {
            gemm128(T, Z, Y, -0.5f, 1.5f, tid);   // T = 1.5 I - 0.5 Z Y
            __syncthreads();
            gemm128(W, Y, T, 1.0f, 0.0f, tid);    // newY = Y T
            __syncthreads();
            gemm128(Y, T, Z, 1.0f, 0.0f, tid);    // newZ = T Z (into old-Y buffer)
            __syncthreads();
            float* nY = W; float* nZ = Y; float* nT = Z; float* nW = T;
            Y = nY; Z = nZ; T = nT; W = nW;
        }
        if (lvl + 1 < NLEV) {                     // restart with Z = I, Y = Y^(1/2)
            for (int idx = tid; idx < MATN * MATN; idx += 256) {
                const int row = idx >> 7, col = idx & 127;
                Z[row * LDSS + col] = (row == col) ? 1.0f : 0.0f;
            }
            __syncthreads();
        }
    }

    // ---- E = I - Y  (into Z) ----
    for (int idx = tid; idx < MATN * MATN; idx += 256) {
        const int row = idx >> 7, col = idx & 127;
        float v = -Y[row * LDSS + col];
        if (row == col) v += 1.0f;
        Z[row * LDSS + col] = v;
    }
    __syncthreads();
    float* E = Z;

    // ---- Horner: P = sum_{j=1..8} E^j / j = E(c1 + E(c2 + ... + E(c7 + c8 E))) ----
    for (int idx = tid; idx < MATN * MATN; idx += 256) {   // S = c8*E + c7*I
        const int row = idx >> 7, col = idx & 127;
        float v = 0.125f * E[row * LDSS + col];
        if (row == col) v += (1.0f / 7.0f);
        T[row * LDSS + col] = v;
    }
    __syncthreads();
    float* S = T;
    float* U = W;
    for (int j = 6; j >= 1; --j) {
        gemm128(U, E, S, 1.0f, 1.0f / (float)j, tid);      // S' = E S + (1/j) I
        __syncthreads();
        float* tmp = S; S = U; U = tmp;
    }
    gemm128(U, E, S, 1.0f, 0.0f, tid);                     // P = E S
    __syncthreads();

    // ---- out = -2^k * P + log(t) * I ----
    const float logt = logf(t);
    for (int idx = tid; idx < MATN * MATN; idx += 256) {
        const int row = idx >> 7, col = idx & 127;
        float v = -64.0f * U[row * LDSS + col];
        if (row == col) v += logt;
        out[base + idx] = v;
    }
}

extern "C" void kernel_launch(void* const* d_in, const int* in_sizes, int n_in,
                              void* d_out, int out_size, void* d_ws, size_t ws_size,
                              hipStream_t stream)
{
    (void)in_sizes; (void)n_in; (void)d_ws; (void)ws_size;
    const float* A   = (const float*)d_in[0];
    float*       out = (float*)d_out;
    const int batches = out_size / (MATN * MATN);   // 4096
    logm_spd_kernel<<<dim3(batches), dim3(256), 0, stream>>>(A, out);
}